// HybridEdgeBuilder_24970939859603
// MI455X (gfx1250) — compile-verified
//
#include <hip/hip_runtime.h>
#include <stdint.h>

typedef __attribute__((ext_vector_type(2))) float v2f;
typedef __attribute__((ext_vector_type(8))) float v8f;

#define N_NODES 4096
#define C_DIM   64
#define KNN     8
#define BATCH   8
#define TILE    16
#define WAVES   8
#define TSTRIDE 20   // 16 + 4 pad: 16B-aligned rows, conflict-free DS stores

// ---------------------------------------------------------------------------
// Phase 1: row-wise L2 normalize (bandwidth bound, float4 vectorized)
// ---------------------------------------------------------------------------
__global__ void normalize_kernel(const float* __restrict__ x,
                                 float* __restrict__ fn, int total_rows) {
    int row = blockIdx.x * blockDim.x + threadIdx.x;
    if (row >= total_rows) return;
    const float4* xr = (const float4*)(x + (size_t)row * C_DIM);
    float4 v[16];
    float s = 0.f;
#pragma unroll
    for (int i = 0; i < 16; ++i) {
        v[i] = xr[i];
        s += v[i].x * v[i].x + v[i].y * v[i].y + v[i].z * v[i].z + v[i].w * v[i].w;
    }
    float inv = 1.0f / fmaxf(sqrtf(s), 1e-12f);
    float4* o = (float4*)(fn + (size_t)row * C_DIM);
#pragma unroll
    for (int i = 0; i < 16; ++i) {
        float4 t = v[i];
        t.x *= inv; t.y *= inv; t.z *= inv; t.w *= inv;
        o[i] = t;
    }
}

// ---------------------------------------------------------------------------
// Guarded sorted-insertion into a descending top-8 list (branchless inner).
// ---------------------------------------------------------------------------
__device__ __forceinline__ void top8_insert(float v, int gidx,
                                            float vals[8], int idx[8]) {
    if (v > vals[7]) {
        float vv = v; int ii = gidx;
#pragma unroll
        for (int s = 0; s < 8; ++s) {
            bool gt = vv > vals[s];
            float tv = gt ? vals[s] : vv;
            int   ti = gt ? idx[s]  : ii;
            vals[s] = gt ? vv : vals[s];
            idx[s]  = gt ? ii : idx[s];
            vv = tv; ii = ti;
        }
    }
}

// ---------------------------------------------------------------------------
// Phase 2: fused sim = fn.fn^T (exact fp32 WMMA 16x16x4) + running top-8.
// 8 waves per 16-row tile; column tiles strided across waves; B operand
// double-buffered in registers; per-wave-private LDS staging tile uses
// hardware in-order DS (s_wait_dscnt) instead of workgroup barriers.
// ---------------------------------------------------------------------------
__launch_bounds__(256)
__global__ void simtopk_kernel(const float* __restrict__ fn,
                               int* __restrict__ topk_idx) {
    __shared__ float tile[WAVES][TILE * TSTRIDE];        // 10 KB
    __shared__ float mvals[TILE][WAVES * 2][KNN];        // 8 KB
    __shared__ int   midx[TILE][WAVES * 2][KNN];         // 8 KB

    const int b       = blockIdx.x / (N_NODES / TILE);
    const int rt      = blockIdx.x % (N_NODES / TILE);
    const int rowBase = rt * TILE;
    const float* fnb  = fn + (size_t)b * N_NODES * C_DIM;

    const int tid  = threadIdx.x;
    const int wave = tid >> 5;
    const int lane = tid & 31;
    const int m    = lane & 15;   // row (A) / col (B) within tile
    const int hi   = lane >> 4;   // lane-group select
    float* wtile   = tile[wave];

    // Preload A: row (rowBase+m), K pairs {4t+2hi, 4t+2hi+1}; reused 32x.
    v2f a[16];
    const float* arow = fnb + (size_t)(rowBase + m) * C_DIM;
#pragma unroll
    for (int t = 0; t < 16; ++t)
        a[t] = *(const v2f*)(arow + 4 * t + 2 * hi);

    float vals[8]; int idx[8];
#pragma unroll
    for (int s = 0; s < 8; ++s) { vals[s] = -INFINITY; idx[s] = 0; }

    const int NT = N_NODES / TILE;   // 256 column tiles, 32 per wave

    // Prime the B double-buffer for this wave's first column tile.
    v2f bcur[16];
    {
        const float* brow = fnb + (size_t)(wave * TILE + m) * C_DIM;
#pragma unroll
        for (int t = 0; t < 16; ++t)
            bcur[t] = *(const v2f*)(brow + 4 * t + 2 * hi);
    }

    for (int ct = wave; ct < NT; ct += WAVES) {
        // Prefetch next column tile while the WMMA chain runs.
        const int ctn = ct + WAVES;
        v2f bnxt[16];
        if (ctn < NT) {
            const float* brow = fnb + (size_t)(ctn * TILE + m) * C_DIM;
#pragma unroll
            for (int t = 0; t < 16; ++t)
                bnxt[t] = *(const v2f*)(brow + 4 * t + 2 * hi);
        }

        v8f c = {};
#pragma unroll
        for (int t = 0; t < 16; ++t)
            c = __builtin_amdgcn_wmma_f32_16x16x4_f32(
                    /*neg_a=*/false, a[t], /*neg_b=*/false, bcur[t],
                    /*c_mod=*/(short)0, c, /*reuse_a=*/false, /*reuse_b=*/false);

        // C/D layout: VGPR j holds (row=j+8*hi, col=m). Stage to wave-private
        // LDS; DS ops of one wave are in order, so a dscnt drain suffices.
#pragma unroll
        for (int j = 0; j < 8; ++j)
            wtile[(j + 8 * hi) * TSTRIDE + m] = c[j];
        asm volatile("s_wait_dscnt 0x0" ::: "memory");

        // lane handles row m, column half [hi*8, hi*8+8): two b128 reads.
        const int colBase = ct * TILE;
        float4 q0 = *(const float4*)(wtile + m * TSTRIDE + hi * 8);
        float4 q1 = *(const float4*)(wtile + m * TSTRIDE + hi * 8 + 4);
        float q[8] = {q0.x, q0.y, q0.z, q0.w, q1.x, q1.y, q1.z, q1.w};
#pragma unroll
        for (int cc = 0; cc < 8; ++cc) {
            int gcol = colBase + hi * 8 + cc;
            float v  = q[cc];
            if (gcol == rowBase + m) v = -INFINITY;   // fill_diagonal_(-inf)
            top8_insert(v, gcol, vals, idx);
        }

        if (ctn < NT) {
#pragma unroll
            for (int t = 0; t < 16; ++t) bcur[t] = bnxt[t];
        }
    }

    // Cross-wave merge: 16 lists (8 waves x 2 half-rows) of 8 per row.
    // Set equality is all that matters: everything is sorted/deduped later.
    const int list = wave * 2 + hi;
#pragma unroll
    for (int s = 0; s < 8; ++s) {
        mvals[m][list][s] = vals[s];
        midx[m][list][s]  = idx[s];
    }
    __syncthreads();

    if (tid < TILE) {
        const int r = tid;
        float fv[8]; int fi[8];
#pragma unroll
        for (int s = 0; s < 8; ++s) { fv[s] = -INFINITY; fi[s] = 0; }
        for (int l = 0; l < WAVES * 2; ++l)
#pragma unroll
            for (int s = 0; s < 8; ++s)
                top8_insert(mvals[r][l][s], midx[r][l][s], fv, fi);
        int* out = topk_idx + ((size_t)b * N_NODES + rowBase + r) * KNN;
#pragma unroll
        for (int s = 0; s < 8; ++s) out[s] = fi[s];
    }
}

// ---------------------------------------------------------------------------
// Phase 3: build padded key arrays  (grid edges ++ knn edges ++ sentinels)
// ---------------------------------------------------------------------------
__global__ void buildkeys_kernel(const int* __restrict__ g, int e_grid,
                                 const int* __restrict__ topk_idx,
                                 uint32_t* __restrict__ keys, int npad) {
    int tid = blockIdx.x * blockDim.x + threadIdx.x;
    int b = tid / npad, e = tid % npad;
    if (b >= BATCH) return;
    uint32_t key;
    if (e < e_grid) {
        key = (uint32_t)g[e] * N_NODES + (uint32_t)g[e_grid + e];
    } else if (e < e_grid + N_NODES * KNN) {
        int t   = e - e_grid;
        int src = t / KNN;
        int tgt = topk_idx[(size_t)b * N_NODES * KNN + t];
        key = (uint32_t)src * N_NODES + (uint32_t)tgt;
    } else {
        key = 0xFFFFFFFFu;  // sentinel sorts after all real keys (< 2^24)
    }
    keys[(size_t)b * npad + e] = key;
}

// ---------------------------------------------------------------------------
// Phase 4: bitonic sort over global memory (one launch per (k,j) stage)
// ---------------------------------------------------------------------------
__global__ void bitonic_kernel(uint32_t* __restrict__ keys, int npad, int k, int j) {
    int tid = blockIdx.x * blockDim.x + threadIdx.x;
    int b = tid / npad, i = tid % npad;
    if (b >= BATCH) return;
    int ixj = i ^ j;
    if (ixj > i) {
        uint32_t* base = keys + (size_t)b * npad;
        uint32_t va = base[i], vb = base[ixj];
        bool up = ((i & k) == 0);
        if ((va > vb) == up) { base[i] = vb; base[ixj] = va; }
    }
}

// ---------------------------------------------------------------------------
// Phase 5: emit edges [B,2,E] (int32) and uniqueness mask [B,E] (u8)
// ---------------------------------------------------------------------------
__global__ void output_kernel(const uint32_t* __restrict__ keys, int npad, int e_total,
                              int* __restrict__ edges, uint8_t* __restrict__ mask) {
    int tid = blockIdx.x * blockDim.x + threadIdx.x;
    int b = tid / e_total, e = tid % e_total;
    if (b >= BATCH) return;
    uint32_t key = keys[(size_t)b * npad + e];
    edges[((size_t)b * 2 + 0) * e_total + e] = (int)(key / N_NODES);
    edges[((size_t)b * 2 + 1) * e_total + e] = (int)(key % N_NODES);
    mask[(size_t)b * e_total + e] =
        (e == 0) ? 1 : (key != keys[(size_t)b * npad + e - 1]);
}

// ---------------------------------------------------------------------------
extern "C" void kernel_launch(void* const* d_in, const int* in_sizes, int n_in,
                              void* d_out, int out_size, void* d_ws, size_t ws_size,
                              hipStream_t stream) {
    const float* x = (const float*)d_in[0];
    const int*   g = (const int*)d_in[1];
    const int e_grid  = in_sizes[1] / 2;                 // 16128
    const int e_total = e_grid + N_NODES * KNN;          // 48896
    int npad = 1; while (npad < e_total) npad <<= 1;     // 65536

    float*    fn   = (float*)d_ws;                                      // 8 MB
    int*      topk = (int*)(fn + (size_t)BATCH * N_NODES * C_DIM);      // 1 MB
    uint32_t* keys = (uint32_t*)(topk + (size_t)BATCH * N_NODES * KNN); // 2 MB

    const int rows = BATCH * N_NODES;
    normalize_kernel<<<(rows + 255) / 256, 256, 0, stream>>>(x, fn, rows);

    simtopk_kernel<<<BATCH * (N_NODES / TILE), 256, 0, stream>>>(fn, topk);

    const int nk = BATCH * npad;
    buildkeys_kernel<<<(nk + 255) / 256, 256, 0, stream>>>(g, e_grid, topk, keys, npad);

    for (int k = 2; k <= npad; k <<= 1)
        for (int j = k >> 1; j > 0; j >>= 1)
            bitonic_kernel<<<(nk + 255) / 256, 256, 0, stream>>>(keys, npad, k, j);

    const int no = BATCH * e_total;
    int*     edges = (int*)d_out;
    uint8_t* mask  = (uint8_t*)((char*)d_out + (size_t)BATCH * 2 * e_total * sizeof(int));
    output_kernel<<<(no + 255) / 256, 256, 0, stream>>>(keys, npad, e_total, edges, mask);
}